// DecoderTASA_13778255086060
// MI455X (gfx1250) — compile-verified
//
#include <hip/hip_runtime.h>
#include <hip/hip_bf16.h>
#include <math.h>

// ---------------------------------------------------------------------------
// DecoderTASA step: GRU cell + logits + log-softmax.  H=1024, V=50257.
// Memory-bound GEMV chain (231 MB of fp32 weights @ 23.3 TB/s ~ 10 us floor).
// GEMVs run through V_WMMA_F32_16X16X4_F32 (exact fp32 matrix pipe), with
// weight tiles staged global->LDS via the gfx1250 async-LDS DMA path when
// available (ASYNCcnt), else coalesced non-temporal b128 loads.
// ---------------------------------------------------------------------------

typedef __attribute__((ext_vector_type(2))) float v2f;
typedef __attribute__((ext_vector_type(4))) float v4f;
typedef __attribute__((ext_vector_type(8))) float v8f;
typedef __attribute__((ext_vector_type(4))) int   v4i;

#define HDIM 1024
#define LDSW_STRIDE 68   // 16B-aligned (68*4=272) and bank-conflict-free for A fetch

#if defined(__AMDGCN__) && __has_builtin(__builtin_amdgcn_global_load_async_to_lds_b128) && __has_builtin(__builtin_amdgcn_s_wait_asynccnt)
#define USE_ASYNC_LDS 1
#else
#define USE_ASYNC_LDS 0
#endif

__device__ __forceinline__ float sigmoidf_(float x) {
    return 1.0f / (1.0f + expf(-x));
}

// --------------------------- x = emb[ix] * prob ----------------------------
__global__ void k_embed(const int* __restrict__ inputs,
                        const float* __restrict__ tau,
                        const float* __restrict__ emb,
                        const float* __restrict__ theta,
                        const float* __restrict__ mu,
                        float* __restrict__ x) {
    int h = blockIdx.x * blockDim.x + threadIdx.x;
    if (h >= HDIM) return;
    int ix = inputs[0];
    float prob = sigmoidf_(theta[ix] + mu[ix] * tau[0]);
    x[h] = emb[(size_t)ix * HDIM + h] * prob;
}

// --------------------- y = W @ v + bias via f32 WMMA -----------------------
// W: R x HDIM row-major. Block = 256 thr = 8 waves; each wave owns one
// 16-row tile (block covers 128 rows). W tiles staged 16x64 into LDS, then
// 16 K=4 WMMA chunks per 64-col block. B operand is v broadcast across the
// 16 columns, so every column of D replicates the 16 dot products.
__global__ __launch_bounds__(256) void k_gemv_wmma(const float* __restrict__ W,
                                                   const float* __restrict__ v,
                                                   const float* __restrict__ bias,
                                                   float* __restrict__ y,
                                                   int R) {
    __shared__ float ldsV[HDIM];
    __shared__ float ldsW[8][16 * LDSW_STRIDE];

    const int tid  = threadIdx.x;
    const int wv   = tid >> 5;
    const int lane = tid & 31;
    const int m    = lane & 15;        // row within tile / N column
    const int half = lane >> 4;        // 0: lanes 0-15, 1: lanes 16-31
    const int rb   = blockIdx.x * 128 + wv * 16;

    for (int h = tid; h < HDIM; h += 256) ldsV[h] = v[h];
    __syncthreads();

    v8f acc = {};
    float* myW = &ldsW[wv][0];

    for (int c = 0; c < HDIM; c += 64) {
        // ---- stage 16 rows x 64 cols of W (coalesced 256B per half-wave) ----
        #pragma unroll
        for (int p = 0; p < 8; ++p) {
            int row = p * 2 + half;
            int gr  = rb + row;
            if (gr > R - 1) gr = R - 1;          // clamp: uniform loop trips
            int col = m * 4;
            const float* src = W + (size_t)gr * HDIM + c + col;
            float* dst = myW + row * LDSW_STRIDE + col;
#if USE_ASYNC_LDS
            // gfx1250 async DMA: global -> LDS without VGPR round-trip
            __builtin_amdgcn_global_load_async_to_lds_b128(
                (v4i*)src, (v4i*)dst, 0, 0);
#else
            *(v4f*)dst = __builtin_nontemporal_load((const v4f*)src);
#endif
        }
#if USE_ASYNC_LDS
        __builtin_amdgcn_s_wait_asynccnt(0);
#endif
        __syncthreads();

        // ---- 16 WMMA K-chunks of 4 ----
        // A layout (32b 16x4): lane holds A[m][2*half], A[m][2*half+1]
        // B layout (4x16) symmetric; broadcast of v across columns.
        #pragma unroll
        for (int kc = 0; kc < 16; ++kc) {
            int k0 = kc * 4 + half * 2;
            v2f a = *(const v2f*)(myW + m * LDSW_STRIDE + k0);
            v2f b = *(const v2f*)(&ldsV[c + k0]);
            acc = __builtin_amdgcn_wmma_f32_16x16x4_f32(
                false, a, false, b, (short)0, acc, false, false);
        }
        __syncthreads();
    }

    // D columns all equal: lane 0 holds y[rb+0..7], lane 16 holds y[rb+8..15]
    if (m == 0) {
        int base = rb + half * 8;
        #pragma unroll
        for (int i = 0; i < 8; ++i) {
            int r = base + i;
            if (r < R) y[r] = acc[i] + bias[r];
        }
    }
}

// ------------------------------- GRU gates ---------------------------------
__global__ void k_gates(const float* __restrict__ gi,
                        const float* __restrict__ gh,
                        const float* __restrict__ hidden,
                        float* __restrict__ h_new) {
    int j = blockIdx.x * blockDim.x + threadIdx.x;
    if (j >= HDIM) return;
    float r = sigmoidf_(gi[j] + gh[j]);
    float z = sigmoidf_(gi[HDIM + j] + gh[HDIM + j]);
    float n = tanhf(gi[2 * HDIM + j] + r * gh[2 * HDIM + j]);
    float h = hidden[j];
    h_new[j] = (1.0f - z) * n + z * h;
}

// ------------------- max + log-sum-exp (single block) ----------------------
__global__ __launch_bounds__(1024) void k_reduce_lse(const float* __restrict__ logits,
                                                     int V, float* __restrict__ red) {
    __shared__ float sm[1024];
    int tid = threadIdx.x;
    float mx = -3.402823466e38f;
    for (int i = tid; i < V; i += 1024) mx = fmaxf(mx, logits[i]);
    sm[tid] = mx;
    __syncthreads();
    for (int s = 512; s > 0; s >>= 1) {
        if (tid < s) sm[tid] = fmaxf(sm[tid], sm[tid + s]);
        __syncthreads();
    }
    float gm = sm[0];
    __syncthreads();
    float sum = 0.0f;
    for (int i = tid; i < V; i += 1024) sum += expf(logits[i] - gm);
    sm[tid] = sum;
    __syncthreads();
    for (int s = 512; s > 0; s >>= 1) {
        if (tid < s) sm[tid] += sm[tid + s];
        __syncthreads();
    }
    if (tid == 0) { red[0] = gm; red[1] = logf(sm[0]); }
}

// --------------------------- out = logits - m - lse ------------------------
__global__ void k_logsoftmax(float* __restrict__ out,
                             const float* __restrict__ red, int V) {
    int i = blockIdx.x * blockDim.x + threadIdx.x;
    if (i < V) out[i] = out[i] - red[0] - red[1];
}

// ---------------------------------------------------------------------------
extern "C" void kernel_launch(void* const* d_in, const int* in_sizes, int n_in,
                              void* d_out, int out_size, void* d_ws, size_t ws_size,
                              hipStream_t stream) {
    const int*   inputs = (const int*)d_in[0];
    const float* hidden = (const float*)d_in[1];   // [1,1,H]
    const float* tau    = (const float*)d_in[2];
    const float* emb    = (const float*)d_in[3];   // [V,H]
    const float* w_ih   = (const float*)d_in[4];   // [3H,H]
    const float* w_hh   = (const float*)d_in[5];   // [3H,H]
    const float* b_ih   = (const float*)d_in[6];   // [3H]
    const float* b_hh   = (const float*)d_in[7];   // [3H]
    const float* w_lin  = (const float*)d_in[8];   // [V,H]
    const float* b_lin  = (const float*)d_in[9];   // [V]
    const float* theta  = (const float*)d_in[10];  // [V]
    const float* mu     = (const float*)d_in[11];  // [V]

    const int H = HDIM;
    const int V = in_sizes[9];                     // b_lin size = V

    float* out   = (float*)d_out;                  // [0,V): log-softmax; [V,V+H): h_new
    float* hnew  = out + V;
    float* ws    = (float*)d_ws;
    float* x     = ws;                             // H
    float* gi    = ws + H;                         // 3H
    float* gh    = ws + 4 * H;                     // 3H
    float* red   = ws + 7 * H;                     // 2

    // 1) scaled embedding
    k_embed<<<(H + 255) / 256, 256, 0, stream>>>(inputs, tau, emb, theta, mu, x);

    // 2,3) GRU input/hidden GEMVs (3072 rows each) via f32 WMMA
    k_gemv_wmma<<<(3 * H + 127) / 128, 256, 0, stream>>>(w_ih, x,      b_ih, gi, 3 * H);
    k_gemv_wmma<<<(3 * H + 127) / 128, 256, 0, stream>>>(w_hh, hidden, b_hh, gh, 3 * H);

    // 4) gate nonlinearities -> h_new (directly into second output slot)
    k_gates<<<(H + 255) / 256, 256, 0, stream>>>(gi, gh, hidden, hnew);

    // 5) logits = w_lin @ h_new + b_lin (dominant: 206 MB stream) via f32 WMMA
    k_gemv_wmma<<<(V + 127) / 128, 256, 0, stream>>>(w_lin, hnew, b_lin, out, V);

    // 6) max + log-sum-exp over V
    k_reduce_lse<<<1, 1024, 0, stream>>>(out, V, red);

    // 7) log-softmax in place
    k_logsoftmax<<<(V + 255) / 256, 256, 0, stream>>>(out, red, V);
}